// CBOW_893353198128
// MI455X (gfx1250) — compile-verified
//
#include <hip/hip_runtime.h>
#include <hip/hip_bf16.h>

// CBOW forward on gfx1250 (MI455X).
// Stage 1: h[N,256] = ReLU( (1/8) * sum_{distinct ctx tokens} W1[:,tok] + b1 )
// Stage 2: out[N,50000] = h @ W2^T + b2 via V_WMMA_F32_16X16X4_F32 (exact fp32),
//          128x128 block tile, 4x2 wave grid, 32x64 wave tile, double-buffered.

#define VOCAB  50000
#define HIDDEN 256
#define NCTX   8      // 2*CTX tokens per example

typedef float v2f __attribute__((ext_vector_type(2)));
typedef float v8f __attribute__((ext_vector_type(8)));

// ---------------------------------------------------------------------------
// Stage 1: embedding gather-sum with SET-semantics dedup, bias, ReLU.
// ---------------------------------------------------------------------------
__global__ __launch_bounds__(HIDDEN) void cbow_hidden_kernel(
    const int*   __restrict__ inputs,  // [N, 8]
    const float* __restrict__ W1,      // [HIDDEN, VOCAB] row-major
    const float* __restrict__ b1,      // [HIDDEN]
    float*       __restrict__ h)       // [N, HIDDEN]
{
    __shared__ int tok[NCTX];
    const int n = blockIdx.x;
    const int t = threadIdx.x;

    if (t < NCTX) tok[t] = inputs[n * NCTX + t];
    __syncthreads();

    float s = 0.0f;
#pragma unroll
    for (int i = 0; i < NCTX; ++i) {
        const int c = tok[i];
        bool dup = false;
#pragma unroll
        for (int j = 0; j < NCTX; ++j) {
            if (j < i) dup = dup || (tok[j] == c);
        }
        if (!dup) s += W1[(size_t)t * VOCAB + c];   // SET semantics: count once
    }
    float v = fmaf(s, 0.125f, b1[t]);
    h[n * HIDDEN + t] = v > 0.0f ? v : 0.0f;
}

// ---------------------------------------------------------------------------
// Stage 2: out = h @ W2^T + b2, fp32 WMMA (16x16x4).
//
// Block = 8 waves as 4(M-groups) x 2(N-groups). Wave tile = 32M x 64N:
//   2 A sub-tiles (reused over 4 N sub-tiles), 4 B sub-tiles (reused over 2 M),
//   8 f32 accumulators (64 VGPRs). Double-buffered K loop (step 4).
// Lane layouts (x4 f32 WMMA):
//   A: lane l -> {A[m0+(l&15), k+2*(l>>4)], +1} (float2, contiguous in K)
//   B: lane l -> {W2[v+(l&15), k+2*(l>>4)], +1} (B = W2^T, rows of W2)
//   D: VGPR j -> D[m0 + j + 8*(l>>4), v + (l&15)]
// ---------------------------------------------------------------------------
__global__ __launch_bounds__(256) void cbow_out_kernel(
    const float* __restrict__ h,    // [N, HIDDEN]
    const float* __restrict__ W2,   // [VOCAB, HIDDEN] row-major
    const float* __restrict__ b2,   // [VOCAB]
    float*       __restrict__ out)  // [N, VOCAB]
{
    const int lane = threadIdx.x & 31;
    const int wave = threadIdx.x >> 5;
    const int wr   = wave >> 1;              // 0..3  (M group)
    const int wc   = wave & 1;               // 0..1  (N group)
    const int lo   = lane & 15;
    const int hi   = lane >> 4;              // selects K pair {2hi, 2hi+1}

    const int m0 = blockIdx.y * 128 + wr * 32;
    const int v0 = blockIdx.x * 128 + wc * 64;

    // Column indices of the 4 N sub-tiles for this lane.
    const int c0 = v0 + lo;
    const int c1 = c0 + 16;
    const int c2 = c0 + 32;
    const int c3 = c0 + 48;
    // Clamp row index for loads past the vocab edge (stores are predicated).
    const int r0 = c0 < VOCAB ? c0 : VOCAB - 1;
    const int r1 = c1 < VOCAB ? c1 : VOCAB - 1;
    const int r2 = c2 < VOCAB ? c2 : VOCAB - 1;
    const int r3 = c3 < VOCAB ? c3 : VOCAB - 1;

    // 32-bit offsets from uniform bases -> saddr-form global loads.
    const int aOff0 = (m0 + lo) * HIDDEN + 2 * hi;        // A sub-tile 0
    const int aOff1 = aOff0 + 16 * HIDDEN;                // A sub-tile 1
    const int bOff0 = r0 * HIDDEN + 2 * hi;
    const int bOff1 = r1 * HIDDEN + 2 * hi;
    const int bOff2 = r2 * HIDDEN + 2 * hi;
    const int bOff3 = r3 * HIDDEN + 2 * hi;

    v8f acc00 = {0.f,0.f,0.f,0.f,0.f,0.f,0.f,0.f};
    v8f acc01 = acc00, acc02 = acc00, acc03 = acc00;
    v8f acc10 = acc00, acc11 = acc00, acc12 = acc00, acc13 = acc00;

    // ---- software-pipelined K loop: load(k+4) before wmma(k) ----
    v2f a0 = *(const v2f*)(h  + aOff0);
    v2f a1 = *(const v2f*)(h  + aOff1);
    v2f b0 = *(const v2f*)(W2 + bOff0);
    v2f b1v = *(const v2f*)(W2 + bOff1);
    v2f b2v = *(const v2f*)(W2 + bOff2);
    v2f b3v = *(const v2f*)(W2 + bOff3);

#pragma unroll 2
    for (int k = 0; k < HIDDEN - 4; k += 4) {
        const v2f a0n = *(const v2f*)(h  + aOff0 + k + 4);
        const v2f a1n = *(const v2f*)(h  + aOff1 + k + 4);
        const v2f b0n = *(const v2f*)(W2 + bOff0 + k + 4);
        const v2f b1n = *(const v2f*)(W2 + bOff1 + k + 4);
        const v2f b2n = *(const v2f*)(W2 + bOff2 + k + 4);
        const v2f b3n = *(const v2f*)(W2 + bOff3 + k + 4);

        acc00 = __builtin_amdgcn_wmma_f32_16x16x4_f32(false, a0, false, b0,  (short)0, acc00, false, false);
        acc10 = __builtin_amdgcn_wmma_f32_16x16x4_f32(false, a1, false, b0,  (short)0, acc10, false, false);
        acc01 = __builtin_amdgcn_wmma_f32_16x16x4_f32(false, a0, false, b1v, (short)0, acc01, false, false);
        acc11 = __builtin_amdgcn_wmma_f32_16x16x4_f32(false, a1, false, b1v, (short)0, acc11, false, false);
        acc02 = __builtin_amdgcn_wmma_f32_16x16x4_f32(false, a0, false, b2v, (short)0, acc02, false, false);
        acc12 = __builtin_amdgcn_wmma_f32_16x16x4_f32(false, a1, false, b2v, (short)0, acc12, false, false);
        acc03 = __builtin_amdgcn_wmma_f32_16x16x4_f32(false, a0, false, b3v, (short)0, acc03, false, false);
        acc13 = __builtin_amdgcn_wmma_f32_16x16x4_f32(false, a1, false, b3v, (short)0, acc13, false, false);

        a0 = a0n; a1 = a1n; b0 = b0n; b1v = b1n; b2v = b2n; b3v = b3n;
    }
    // tail k = HIDDEN-4
    acc00 = __builtin_amdgcn_wmma_f32_16x16x4_f32(false, a0, false, b0,  (short)0, acc00, false, false);
    acc10 = __builtin_amdgcn_wmma_f32_16x16x4_f32(false, a1, false, b0,  (short)0, acc10, false, false);
    acc01 = __builtin_amdgcn_wmma_f32_16x16x4_f32(false, a0, false, b1v, (short)0, acc01, false, false);
    acc11 = __builtin_amdgcn_wmma_f32_16x16x4_f32(false, a1, false, b1v, (short)0, acc11, false, false);
    acc02 = __builtin_amdgcn_wmma_f32_16x16x4_f32(false, a0, false, b2v, (short)0, acc02, false, false);
    acc12 = __builtin_amdgcn_wmma_f32_16x16x4_f32(false, a1, false, b2v, (short)0, acc12, false, false);
    acc03 = __builtin_amdgcn_wmma_f32_16x16x4_f32(false, a0, false, b3v, (short)0, acc03, false, false);
    acc13 = __builtin_amdgcn_wmma_f32_16x16x4_f32(false, a1, false, b3v, (short)0, acc13, false, false);

    // ---- epilogue: bias + predicated stores ----
    const float bias0 = b2[r0];
    const float bias1 = b2[r1];
    const float bias2 = b2[r2];
    const float bias3 = b2[r3];

#pragma unroll
    for (int j = 0; j < 8; ++j) {
        const int mrow = m0 + j + 8 * hi;
        float* orowA = out + (size_t)mrow * VOCAB;          // A sub-tile 0
        float* orowB = orowA + (size_t)16 * VOCAB;          // A sub-tile 1
        if (c0 < VOCAB) { orowA[c0] = acc00[j] + bias0; orowB[c0] = acc10[j] + bias0; }
        if (c1 < VOCAB) { orowA[c1] = acc01[j] + bias1; orowB[c1] = acc11[j] + bias1; }
        if (c2 < VOCAB) { orowA[c2] = acc02[j] + bias2; orowB[c2] = acc12[j] + bias2; }
        if (c3 < VOCAB) { orowA[c3] = acc03[j] + bias3; orowB[c3] = acc13[j] + bias3; }
    }
}

// ---------------------------------------------------------------------------
extern "C" void kernel_launch(void* const* d_in, const int* in_sizes, int n_in,
                              void* d_out, int out_size, void* d_ws, size_t ws_size,
                              hipStream_t stream)
{
    const int*   inputs = (const int*)  d_in[0];  // [N, 8]
    const float* W1     = (const float*)d_in[1];  // [256, 50000]
    const float* b1     = (const float*)d_in[2];  // [256]
    const float* W2     = (const float*)d_in[3];  // [50000, 256]
    const float* b2     = (const float*)d_in[4];  // [50000]
    float*       out    = (float*)      d_out;    // [N, 50000]

    const int N = in_sizes[0] / NCTX;             // 2048
    float* hbuf = (float*)d_ws;                   // N*HIDDEN*4 = 2 MB scratch

    cbow_hidden_kernel<<<dim3(N), dim3(HIDDEN), 0, stream>>>(inputs, W1, b1, hbuf);

    const int gx = (VOCAB + 127) / 128;           // 391
    const int gy = N / 128;                       // 16
    cbow_out_kernel<<<dim3(gx, gy), dim3(256), 0, stream>>>(hbuf, W2, b2, out);
}